// SMSLayer_6279242187229
// MI455X (gfx1250) — compile-verified
//
#include <hip/hip_runtime.h>
#include <hip/hip_bf16.h>

// ---------------------------------------------------------------------------
// FEM assemble + dense LU solve for MI455X (gfx1250, wave32).
//   n = 6144 dofs, M = 10240 tets.  K (6144^2 fp32 = 151MB) fits in 192MB L2.
//   Matrix math: V_WMMA_F32_16X16X4_F32.  B-panel staging: TDM
//   tensor_load_to_lds (pad 8 DWORDs per 64 -> LDS row stride 72, fully
//   bank-conflict-free for the half-wave-split WMMA B reads) + s_wait_tensorcnt.
// Workspace layout (d_ws): K[n*n] | y[n] | bc[n]   (~151.05 MB)
// ---------------------------------------------------------------------------

typedef __attribute__((ext_vector_type(2)))  float        v2f;
typedef __attribute__((ext_vector_type(8)))  float        v8f;
typedef __attribute__((ext_vector_type(4)))  unsigned int v4u;
typedef __attribute__((ext_vector_type(4)))  int          v4i;
typedef __attribute__((ext_vector_type(8)))  int          v8i;

#define NDOF     6144
#define NNODES   2048
#define MTETS    10240
#define NB       64          // LU block size (K depth of trailing update)
#define NSTEP    (NDOF / NB) // 96
#define LOG_A_MIN 6.2146080984f   // log(500)
#define LOG_A_MAX 9.2103403720f   // log(10000)

#if defined(__has_builtin)
#if __has_builtin(__builtin_amdgcn_tensor_load_to_lds)
#define HAVE_TDM 1
#endif
#endif

// ---------------------------------------------------------------------------
// Per-element G-row fetch: rows 0-2 = r_axis, 3-5 = r_shear, 6 = r_vol, 7 = 0
__device__ __forceinline__ float grow_f(const float* ra, const float* rs,
                                        const float* rv, int k, int i) {
  if (k < 3) return ra[k * 12 + i];
  if (k < 6) return rs[(k - 3) * 12 + i];
  if (k == 6) return rv[i];
  return 0.0f;
}

// One wave per element: Ke = vol*(alpha*Ka + 2alpha*Ks + 8alpha*Kv) = (G^T S) G
// computed as two V_WMMA_F32_16X16X4_F32 (K = 0..3, 4..7), then atomic scatter.
__global__ void assemble_ke(const float* __restrict__ theta,
                            const float* __restrict__ delta,
                            const float* __restrict__ r_axis,
                            const float* __restrict__ r_shear,
                            const float* __restrict__ r_vol,
                            const float* __restrict__ vol,
                            const int*   __restrict__ tets,
                            float* __restrict__ K) {
  const int wave = (blockIdx.x * blockDim.x + threadIdx.x) >> 5;
  const int lane = threadIdx.x & 31;
  if (wave >= MTETS) return;  // wave-uniform

  float la = theta[0] + delta[wave];
  la = fminf(fmaxf(la, LOG_A_MIN), LOG_A_MAX);
  const float alpha = __expf(la);
  const float v  = vol[wave];
  const float ca = v * alpha;          // axis rows 0..2
  const float cb = v * 2.0f * alpha;   // shear rows 3..5
  const float ck = v * 8.0f * alpha;   // volumetric row 6 (4*nu/(1-2nu)=8)

  const float* ra = r_axis  + (size_t)wave * 36;
  const float* rs = r_shear + (size_t)wave * 36;
  const float* rv = r_vol   + (size_t)wave * 12;

  const int i = lane & 15;
  const int g = lane >> 4;
  const bool valid = (i < 12);

  v8f c = {};
  {  // K-chunk 0: rows {0,1} lanes 0-15, {2,3} lanes 16-31
    const int k0 = 2 * g;
    const float s0 = ca;
    const float s1 = (g == 0) ? ca : cb;
    v2f a, b;
    a[0] = valid ? s0 * grow_f(ra, rs, rv, k0,     i) : 0.0f;
    a[1] = valid ? s1 * grow_f(ra, rs, rv, k0 + 1, i) : 0.0f;
    b[0] = valid ?      grow_f(ra, rs, rv, k0,     i) : 0.0f;
    b[1] = valid ?      grow_f(ra, rs, rv, k0 + 1, i) : 0.0f;
    c = __builtin_amdgcn_wmma_f32_16x16x4_f32(false, a, false, b,
                                              (short)0, c, false, false);
  }
  {  // K-chunk 1: rows {4,5} lanes 0-15, {6,7} lanes 16-31
    const int k0 = 4 + 2 * g;
    const float s0 = (g == 0) ? cb : ck;
    const float s1 = (g == 0) ? cb : 0.0f;
    v2f a, b;
    a[0] = valid ? s0 * grow_f(ra, rs, rv, k0,     i) : 0.0f;
    a[1] = valid ? s1 * grow_f(ra, rs, rv, k0 + 1, i) : 0.0f;
    b[0] = valid ?      grow_f(ra, rs, rv, k0,     i) : 0.0f;
    b[1] = valid ?      grow_f(ra, rs, rv, k0 + 1, i) : 0.0f;
    c = __builtin_amdgcn_wmma_f32_16x16x4_f32(false, a, false, b,
                                              (short)0, c, false, false);
  }

  if (valid) {
    const int* tet = tets + wave * 4;
    const int cd = 3 * tet[i / 3] + (i % 3);
    const int rowbase = g * 8;
#pragma unroll
    for (int r = 0; r < 8; ++r) {
      const int row = rowbase + r;
      if (row < 12) {
        const int rd = 3 * tet[row / 3] + (row % 3);
        atomicAdd(&K[(size_t)rd * NDOF + cd], c[r]);
      }
    }
  }
}

// ---------------------------------------------------------------------------
__global__ void apply_bc(float* __restrict__ K, const int* __restrict__ bnd) {
  const int idx = blockIdx.x * 256 + threadIdx.x;   // n*n = 37.7M < 2^31
  const int i = idx / NDOF;
  const int j = idx - i * NDOF;
  if (bnd[i / 3] | bnd[j / 3]) K[idx] = (i == j) ? 1.0f : 0.0f;
}

__global__ void make_rhs(const float* __restrict__ b,
                         const int* __restrict__ bnd,
                         float* __restrict__ bc) {
  const int i = blockIdx.x * 256 + threadIdx.x;
  if (i < NDOF) bc[i] = bnd[i / 3] ? 0.0f : b[i];
}

// ---------------------------------------------------------------------------
__global__ void lu_diag(float* __restrict__ K, int kb) {
  __shared__ float D[NB * NB];
  const int tid = threadIdx.x;
  for (int t = tid; t < NB * NB; t += 256)
    D[t] = K[(size_t)(kb + t / NB) * NDOF + kb + (t & (NB - 1))];
  __syncthreads();
  for (int p = 0; p < NB - 1; ++p) {
    const float dp = D[p * NB + p];
    for (int i = p + 1 + tid; i < NB; i += 256) D[i * NB + p] /= dp;
    __syncthreads();
    const int w = NB - 1 - p;
    for (int t = tid; t < w * w; t += 256) {
      const int i = p + 1 + t / w;
      const int j = p + 1 + t % w;
      D[i * NB + j] -= D[i * NB + p] * D[p * NB + j];
    }
    __syncthreads();
  }
  for (int t = tid; t < NB * NB; t += 256)
    K[(size_t)(kb + t / NB) * NDOF + kb + (t & (NB - 1))] = D[t];
}

// U panel: one column per thread, unit-lower forward substitution in registers.
__global__ void lu_row_panel(float* __restrict__ K, int kb, int rem) {
  __shared__ float Ls[NB * NB];
  const int tid = threadIdx.x;
  for (int t = tid; t < NB * NB; t += 256)
    Ls[t] = K[(size_t)(kb + t / NB) * NDOF + kb + (t & (NB - 1))];
  __syncthreads();
  const int j = kb + NB + blockIdx.x * blockDim.x + tid;
  if (j >= NDOF) return;
  float x[NB];
#pragma unroll
  for (int p = 0; p < NB; ++p) {
    float acc = K[(size_t)(kb + p) * NDOF + j];
#pragma unroll
    for (int q = 0; q < p; ++q) acc -= Ls[p * NB + q] * x[q];
    x[p] = acc;
    K[(size_t)(kb + p) * NDOF + j] = acc;
  }
}

// L panel: one row per thread.
__global__ void lu_col_panel(float* __restrict__ K, int kb, int rem) {
  __shared__ float Us[NB * NB];
  const int tid = threadIdx.x;
  for (int t = tid; t < NB * NB; t += 256)
    Us[t] = K[(size_t)(kb + t / NB) * NDOF + kb + (t & (NB - 1))];
  __syncthreads();
  const int i = kb + NB + blockIdx.x * blockDim.x + tid;
  if (i >= NDOF) return;
  float x[NB];
#pragma unroll
  for (int p = 0; p < NB; ++p) {
    float acc = K[(size_t)i * NDOF + kb + p];
#pragma unroll
    for (int q = 0; q < p; ++q) acc -= x[q] * Us[q * NB + p];
    acc /= Us[p * NB + p];
    x[p] = acc;
    K[(size_t)i * NDOF + kb + p] = acc;
  }
}

// ---------------------------------------------------------------------------
// Trailing update.  Block = 8 waves computes a 128x64 patch of C -= A*B.
// B panel (64x64 of the U block-row) staged into LDS once per block by the
// Tensor Data Mover (8 pad DWORDs per 64 -> LDS row stride 72; with the
// half-wave row split (k, k+2) a stride s is conflict-free iff
// 2s mod 64 in [16,48]; 144 mod 64 = 16 -> all B ds_loads conflict-free).
// Each wave: 16x64 strip, 4 accumulators, A pair reused 4x ->
// 64 chained V_WMMA_F32_16X16X4_F32 per wave.
#define BS_STRIDE 72

__device__ __forceinline__ void stage_b_panel(float* __restrict__ Bs,
                                              const float* __restrict__ K,
                                              int kb, int tj0) {
#ifdef HAVE_TDM
  if ((threadIdx.x >> 5) == 0) {            // wave 0 issues the DMA
    const unsigned long long gaddr =
        (unsigned long long)(const void*)&K[(size_t)kb * NDOF + tj0];
    const unsigned ldsOff = (unsigned)(unsigned long long)(const void*)Bs;

    // D# group 0: count=1 | lds_addr | global_addr | type=2
    v4u g0;
    g0[0] = 1u;                                         // count=1, flags 0
    g0[1] = ldsOff;                                     // LDS byte address
    g0[2] = (unsigned)(gaddr & 0xFFFFFFFFull);          // global addr [95:64]
    g0[3] = (unsigned)((gaddr >> 32) & 0x01FFFFFFull)   // global addr [120:96]
            | 0x80000000u;                              // type = 2 (image)
    // D# group 1: data_size=4B, pad 8 DWORDs per 64, tile 64x64, stride 6144
    v8i g1;
    g1[0] = (int)((2u << 16)      // data_size = 4 bytes
                | (1u << 20)      // pad_enable
                | (5u << 22)      // pad_interval: 64 DWORDs
                | (7u << 25));    // pad_amount: 8 DWORDs
    g1[1] = (int)(64u << 16);     // tensor_dim0 = 64 (low 16)
    g1[2] = (int)(64u << 16);     // tensor_dim0 hi = 0 | tensor_dim1 = 64
    g1[3] = (int)(64u << 16);     // tensor_dim1 hi = 0 | tile_dim0 = 64
    g1[4] = 64;                   // tile_dim1 = 64, tile_dim2 = 0
    g1[5] = NDOF;                 // tensor_dim0_stride = 6144 (low 32)
    g1[6] = 0;
    g1[7] = 0;
    v4i gz = {};                  // groups 2/3 unused (2D tensor)
#if __has_include(<hip/amd_detail/amd_gfx1250_TDM.h>)
    v8i gz8 = {};
    __builtin_amdgcn_tensor_load_to_lds(g0, g1, gz, gz, gz8, 0);
#else
    __builtin_amdgcn_tensor_load_to_lds(g0, g1, gz, gz, 0);
#endif
    __builtin_amdgcn_s_wait_tensorcnt(0);
  }
#else
  for (int t = threadIdx.x; t < NB * 64; t += 256) {
    const int r = t >> 6, cc = t & 63;
    Bs[r * BS_STRIDE + cc] = K[(size_t)(kb + r) * NDOF + tj0 + cc];
  }
#endif
  __syncthreads();
}

__global__ void lu_trailing(float* __restrict__ K, int kb, int rem) {
  __shared__ float Bs[NB * BS_STRIDE];    // 64 rows, padded stride 72 (18 KB)
  const int lane = threadIdx.x & 31;
  const int wv   = threadIdx.x >> 5;
  const int base = kb + NB;
  const int tj0  = base + blockIdx.x * 64;

  stage_b_panel(Bs, K, kb, tj0);          // all waves participate in barrier

  const int tIdx = blockIdx.y * 8 + wv;   // 16-row strip index
  if (tIdx * 16 >= rem) return;           // wave-uniform, after the barrier
  const int ti0 = base + tIdx * 16;

  const int col = lane & 15;
  const int g   = lane >> 4;
  const int rowbase = g * 8;              // C/D rows: 0-7 lo half, 8-15 hi half

  v8f c0, c1, c2, c3;
#pragma unroll
  for (int r = 0; r < 8; ++r) {
    const size_t ro = (size_t)(ti0 + rowbase + r) * NDOF + tj0 + col;
    c0[r] = K[ro];
    c1[r] = K[ro + 16];
    c2[r] = K[ro + 32];
    c3[r] = K[ro + 48];
  }

  const int arow = ti0 + col;             // A row for this lane
#pragma unroll
  for (int kk = 0; kk < 16; ++kk) {
    const int krel = kk * 4 + 2 * g;      // K pair: {0,1} lo half, {2,3} hi
    v2f a, b0, b1, b2, b3;
    a[0]  = -K[(size_t)arow * NDOF + kb + krel];
    a[1]  = -K[(size_t)arow * NDOF + kb + krel + 1];
    const int r0 = krel * BS_STRIDE + col;
    const int r1 = (krel + 1) * BS_STRIDE + col;
    b0[0] = Bs[r0];      b0[1] = Bs[r1];
    b1[0] = Bs[r0 + 16]; b1[1] = Bs[r1 + 16];
    b2[0] = Bs[r0 + 32]; b2[1] = Bs[r1 + 32];
    b3[0] = Bs[r0 + 48]; b3[1] = Bs[r1 + 48];
    c0 = __builtin_amdgcn_wmma_f32_16x16x4_f32(false, a, false, b0,
                                               (short)0, c0, false, false);
    c1 = __builtin_amdgcn_wmma_f32_16x16x4_f32(false, a, false, b1,
                                               (short)0, c1, false, false);
    c2 = __builtin_amdgcn_wmma_f32_16x16x4_f32(false, a, false, b2,
                                               (short)0, c2, false, false);
    c3 = __builtin_amdgcn_wmma_f32_16x16x4_f32(false, a, false, b3,
                                               (short)0, c3, false, false);
  }
#pragma unroll
  for (int r = 0; r < 8; ++r) {
    const size_t ro = (size_t)(ti0 + rowbase + r) * NDOF + tj0 + col;
    K[ro]      = c0[r];
    K[ro + 16] = c1[r];
    K[ro + 32] = c2[r];
    K[ro + 48] = c3[r];
  }
}

// ---------------------------------------------------------------------------
__global__ void fwd_solve(const float* __restrict__ K,
                          const float* __restrict__ bc,
                          float* __restrict__ y) {
  __shared__ float red[256];
  const int tid = threadIdx.x;
  for (int i = 0; i < NDOF; ++i) {
    float s = 0.0f;
    for (int j = tid; j < i; j += 256) s += K[(size_t)i * NDOF + j] * y[j];
    red[tid] = s;
    __syncthreads();
    for (int off = 128; off > 0; off >>= 1) {
      if (tid < off) red[tid] += red[tid + off];
      __syncthreads();
    }
    if (tid == 0) y[i] = bc[i] - red[0];
    __syncthreads();
  }
}

__global__ void bwd_solve(const float* __restrict__ K,
                          const float* __restrict__ y,
                          float* __restrict__ u) {
  __shared__ float red[256];
  const int tid = threadIdx.x;
  for (int i = NDOF - 1; i >= 0; --i) {
    float s = 0.0f;
    for (int j = i + 1 + tid; j < NDOF; j += 256)
      s += K[(size_t)i * NDOF + j] * u[j];
    red[tid] = s;
    __syncthreads();
    for (int off = 128; off > 0; off >>= 1) {
      if (tid < off) red[tid] += red[tid + off];
      __syncthreads();
    }
    if (tid == 0) u[i] = (y[i] - red[0]) / K[(size_t)i * NDOF + i];
    __syncthreads();
  }
}

// ---------------------------------------------------------------------------
extern "C" void kernel_launch(void* const* d_in, const int* in_sizes, int n_in,
                              void* d_out, int out_size, void* d_ws, size_t ws_size,
                              hipStream_t stream) {
  (void)in_sizes; (void)n_in; (void)out_size; (void)ws_size;
  const float* theta   = (const float*)d_in[0];
  const float* delta   = (const float*)d_in[1];
  const float* r_axis  = (const float*)d_in[2];
  const float* r_shear = (const float*)d_in[3];
  const float* r_vol   = (const float*)d_in[4];
  const float* vol     = (const float*)d_in[5];
  const float* b       = (const float*)d_in[6];
  const int*   tets    = (const int*)d_in[7];
  const int*   bnd     = (const int*)d_in[8];
  float* u = (float*)d_out;

  float* K  = (float*)d_ws;                       // n*n fp32, L2-resident
  float* y  = K + (size_t)NDOF * NDOF;
  float* bc = y + NDOF;

  hipMemsetAsync(K, 0, (size_t)NDOF * NDOF * sizeof(float), stream);

  assemble_ke<<<MTETS / 8, 256, 0, stream>>>(theta, delta, r_axis, r_shear,
                                             r_vol, vol, tets, K);
  apply_bc<<<(NDOF * NDOF) / 256, 256, 0, stream>>>(K, bnd);
  make_rhs<<<(NDOF + 255) / 256, 256, 0, stream>>>(b, bnd, bc);

  for (int k = 0; k < NSTEP; ++k) {
    const int kb = k * NB;
    lu_diag<<<1, 256, 0, stream>>>(K, kb);
    const int rem = NDOF - kb - NB;
    if (rem > 0) {
      lu_row_panel<<<(rem + 255) / 256, 256, 0, stream>>>(K, kb, rem);
      lu_col_panel<<<(rem + 255) / 256, 256, 0, stream>>>(K, kb, rem);
      dim3 grid(rem / 64, (rem + 127) / 128);   // rem is a multiple of 64
      lu_trailing<<<grid, 256, 0, stream>>>(K, kb, rem);
    }
  }

  fwd_solve<<<1, 256, 0, stream>>>(K, bc, y);
  bwd_solve<<<1, 256, 0, stream>>>(K, y, u);
}